// LocalizeLoss_Anchor_78512002171323
// MI455X (gfx1250) — compile-verified
//
#include <hip/hip_runtime.h>
#include <hip/hip_bf16.h>

typedef __attribute__((ext_vector_type(2))) float v2f;
typedef __attribute__((ext_vector_type(8))) float v8f;

#define LOSS_EPS 1e-7f
// 4 / pi^2
#define INV_PI2_X4 0.40528473456935108577f

// Full 32-lane wave sum using V_WMMA_F32_16X16X4_F32.
// A (16x4): a0 = s -> A[m,0]=s[m] (lanes 0-15), A[m,2]=s[m+16] (lanes 16-31); a1=0.
// B (4x16) = ones. D[m,n] = s[m] + s[m+16] for all n.
// D VGPR j: lanes 0-15 hold row M=j, lanes 16-31 hold row M=j+8.
// Sum of the 8 accumulators per lane + one xor-16 shuffle = full wave sum in every lane.
__device__ __forceinline__ float wave_reduce_wmma(float s) {
    v2f a; a[0] = s;    a[1] = 0.0f;
    v2f b; b[0] = 1.0f; b[1] = 1.0f;
    v8f c = {};
    c = __builtin_amdgcn_wmma_f32_16x16x4_f32(
        /*neg_a=*/false, a, /*neg_b=*/false, b,
        /*c_mod=*/(short)0, c, /*reuse_a=*/false, /*reuse_b=*/false);
    float t = c[0] + c[1] + c[2] + c[3] + c[4] + c[5] + c[6] + c[7];
    t += __shfl_xor(t, 16, 32);
    return t;
}

__global__ __launch_bounds__(256) void loss_partial_kernel(
    const float* __restrict__ pred,   // (B, 30)
    const float* __restrict__ tgt,    // (B, 5)
    float* __restrict__ partial,      // (gridDim.x)
    int nrows)
{
    const int tid    = blockIdx.x * blockDim.x + threadIdx.x;
    const int stride = gridDim.x * blockDim.x;

    float acc     = 0.0f;  // l_bbox + l_conf contributions
    float acc_cls = 0.0f;  // sum of picked log-probs (negated/scaled at the end)

    for (int r = tid; r < nrows; r += stride) {
        // ---- targets (streamed once: NT loads) ----
        const float* t5 = tgt + (size_t)r * 5;
        float lf = __builtin_nontemporal_load(t5 + 0);
        float tx = __builtin_nontemporal_load(t5 + 1);
        float ty = __builtin_nontemporal_load(t5 + 2);
        float tw = __builtin_nontemporal_load(t5 + 3);
        float th = __builtin_nontemporal_load(t5 + 4);
        int l = (int)lf;
        l = (l < 0) ? 0 : ((l > 4) ? 4 : l);

        // Hoist anchor-independent target terms (fast divide: tolerance >> 2-3 ulp)
        float at2   = atanf(__fdividef(tw, th + LOSS_EPS));
        float b2x1  = tx - tw * 0.5f, b2x2 = tx + tw * 0.5f;
        float b2y1  = ty - th * 0.5f, b2y2 = ty + th * 0.5f;
        float area2 = tw * th;

        // predictions row: 120 B, 8-byte aligned -> b64 NT loads
        const v2f* p2 = reinterpret_cast<const v2f*>(pred + (size_t)r * 30);

        #pragma unroll
        for (int aIdx = 0; aIdx < 3; ++aIdx) {
            v2f q0 = __builtin_nontemporal_load(p2 + aIdx * 5 + 0);
            v2f q1 = __builtin_nontemporal_load(p2 + aIdx * 5 + 1);
            v2f q2 = __builtin_nontemporal_load(p2 + aIdx * 5 + 2);
            v2f q3 = __builtin_nontemporal_load(p2 + aIdx * 5 + 3);
            v2f q4 = __builtin_nontemporal_load(p2 + aIdx * 5 + 4);

            float x0 = q0[0], x1 = q0[1], x2 = q1[0], x3 = q1[1], x4 = q2[0];
            float conf = q2[1];
            float bx = q3[0], by = q3[1], bw = q4[0], bh = q4[1];

            // ---- l_cls: log_softmax pick ----
            float m  = fmaxf(fmaxf(fmaxf(x0, x1), fmaxf(x2, x3)), x4);
            float se = __expf(x0 - m) + __expf(x1 - m) + __expf(x2 - m)
                     + __expf(x3 - m) + __expf(x4 - m);
            float lse = m + __logf(se);
            float xl  = (l == 0) ? x0 : (l == 1) ? x1 : (l == 2) ? x2
                      : (l == 3) ? x3 : x4;
            acc_cls += xl - lse;

            // ---- l_bbox ----
            float dx = bx - tx, dy = by - ty, dw = bw - tw, dh = bh - th;
            acc += dx * dx + dy * dy + dw * dw + dh * dh;

            // ---- CIoU (forward only; stop_gradient in reference) ----
            float b1x1 = bx - bw * 0.5f, b1x2 = bx + bw * 0.5f;
            float b1y1 = by - bh * 0.5f, b1y2 = by + bh * 0.5f;
            float iw   = fmaxf(fminf(b1x2, b2x2) - fmaxf(b1x1, b2x1), 0.0f);
            float ih   = fmaxf(fminf(b1y2, b2y2) - fmaxf(b1y1, b2y1), 0.0f);
            float inter = iw * ih;
            float uni   = bw * bh + area2 - inter + LOSS_EPS;
            float iou   = __fdividef(inter, uni);
            float cw = fmaxf(b1x2, b2x2) - fminf(b1x1, b2x1);
            float ch = fmaxf(b1y2, b2y2) - fminf(b1y1, b2y1);
            float c2 = cw * cw + ch * ch + LOSS_EPS;
            // ((b2x1+b2x2-b1x1-b1x2)^2 + (..y..)^2)/4 == (tx-bx)^2 + (ty-by)^2
            float rho2  = dx * dx + dy * dy;
            float dat   = at2 - atanf(__fdividef(bw, bh + LOSS_EPS));
            float v     = INV_PI2_X4 * dat * dat;
            // v * alpha = v^2 / (v - iou + 1 + eps)
            float valpha = __fdividef(v * v, v - iou + (1.0f + LOSS_EPS));
            float ciou   = iou - (__fdividef(rho2, c2) + valpha);

            // ---- l_conf ----
            float e = ciou - conf;
            acc += e * e;
        }
    }

    // l_cls = -(1/B) * sum(picked); fold the linear scale per thread.
    float part = acc - acc_cls * (1.0f / (float)nrows);

    // Wave-level sum via WMMA (EXEC is all-ones here), then fixed-order block sum.
    float wsum = wave_reduce_wmma(part);

    __shared__ float smem[8];
    int lane = threadIdx.x & 31;
    int wid  = threadIdx.x >> 5;
    if (lane == 0) smem[wid] = wsum;
    __syncthreads();
    if (threadIdx.x == 0) {
        float s = 0.0f;
        #pragma unroll
        for (int i = 0; i < 8; ++i) s += smem[i];
        partial[blockIdx.x] = s;
    }
}

__global__ __launch_bounds__(256) void final_reduce_kernel(
    const float* __restrict__ partial, float* __restrict__ out, int n)
{
    __shared__ float smem[256];
    float s = 0.0f;
    for (int i = threadIdx.x; i < n; i += 256) s += partial[i];
    smem[threadIdx.x] = s;
    __syncthreads();
    #pragma unroll
    for (int off = 128; off > 0; off >>= 1) {
        if ((int)threadIdx.x < off) smem[threadIdx.x] += smem[threadIdx.x + off];
        __syncthreads();
    }
    if (threadIdx.x == 0) out[0] = smem[0];
}

extern "C" void kernel_launch(void* const* d_in, const int* in_sizes, int n_in,
                              void* d_out, int out_size, void* d_ws, size_t ws_size,
                              hipStream_t stream) {
    const float* pred = (const float*)d_in[0];
    const float* tgt  = (const float*)d_in[1];
    int nrows = in_sizes[0] / 30;

    int nblk = 2048;  // 2048 blocks * 8 wave32 = 16K waves; 4 rows/thread
    if ((size_t)nblk * sizeof(float) > ws_size) {
        nblk = (int)(ws_size / sizeof(float));
        if (nblk < 1) nblk = 1;
    }
    float* partial = (float*)d_ws;

    loss_partial_kernel<<<nblk, 256, 0, stream>>>(pred, tgt, partial, nrows);
    final_reduce_kernel<<<1, 256, 0, stream>>>(partial, (float*)d_out, nblk);
}